// GCNBaseline_32538672234673
// MI455X (gfx1250) — compile-verified
//
#include <hip/hip_runtime.h>

#define NN 100000
#define NE 1600000
#define NG 512
#define DD 128
#define NC 10

typedef float v2f __attribute__((ext_vector_type(2)));
typedef float v8f __attribute__((ext_vector_type(8)));

// ---------------------------------------------------------------- utilities
__global__ void zero_f32(float* __restrict__ p, int n4) {
    int i = blockIdx.x * blockDim.x + threadIdx.x;
    if (i < n4) {
        float4 z = make_float4(0.f, 0.f, 0.f, 0.f);
        *(float4*)(p + (size_t)i * 4) = z;
    }
}

__global__ void deg_count(const int* __restrict__ dst, int* __restrict__ deg, int nE) {
    int e = blockIdx.x * blockDim.x + threadIdx.x;
    if (e < nE) atomicAdd(&deg[dst[e]], 1);
}

__global__ void make_dinv(const int* __restrict__ deg, float* __restrict__ dinv, int n) {
    int i = blockIdx.x * blockDim.x + threadIdx.x;
    if (i < n) dinv[i] = rsqrtf((float)deg[i] + 1.0f);   // +1 self loop -> finite
}

// ------------------------------------------------------- exclusive scan (CSR)
// k1: per-block (256-wide) exclusive scan, block totals out.
__global__ void scan_block(const int* __restrict__ in, int* __restrict__ outEx,
                           int* __restrict__ bsum, int n) {
    __shared__ int s[256];
    int gid = blockIdx.x * 256 + threadIdx.x;
    int v = (gid < n) ? in[gid] : 0;
    s[threadIdx.x] = v;
    __syncthreads();
    for (int off = 1; off < 256; off <<= 1) {
        int t = (threadIdx.x >= off) ? s[threadIdx.x - off] : 0;
        __syncthreads();
        s[threadIdx.x] += t;
        __syncthreads();
    }
    if (gid < n) outEx[gid] = s[threadIdx.x] - v;
    if (threadIdx.x == 255) bsum[blockIdx.x] = s[255];
}

// k2: single-block exclusive scan of block sums (nb <= 512), in place.
__global__ void scan_sums(int* __restrict__ bsum, int nb) {
    __shared__ int s[512];
    int v = ((int)threadIdx.x < nb) ? bsum[threadIdx.x] : 0;
    s[threadIdx.x] = v;
    __syncthreads();
    for (int off = 1; off < 512; off <<= 1) {
        int t = (threadIdx.x >= off) ? s[threadIdx.x - off] : 0;
        __syncthreads();
        s[threadIdx.x] += t;
        __syncthreads();
    }
    if ((int)threadIdx.x < nb) bsum[threadIdx.x] = s[threadIdx.x] - v;
}

// k3: add block offsets; set rowptr[n] = total.
__global__ void add_offsets(int* __restrict__ rowptr, const int* __restrict__ bsum,
                            int n, int total) {
    int gid = blockIdx.x * 256 + threadIdx.x;
    if (gid < n) rowptr[gid] += bsum[gid >> 8];
    if (gid == 0) rowptr[n] = total;
}

__global__ void csr_fill(const int* __restrict__ src, const int* __restrict__ dst,
                         const int* __restrict__ rowptr, int* __restrict__ cursor,
                         int* __restrict__ colsrc, int nE) {
    int e = blockIdx.x * blockDim.x + threadIdx.x;
    if (e >= nE) return;
    int d = dst[e];
    int pos = rowptr[d] + atomicAdd(&cursor[d], 1);
    colsrc[pos] = src[e];
}

// ------------------------------------------------- WMMA f32 GEMM (K = N = 128)
// out[M x 128] = A[M x 128] @ W[128 x 128] (+ bias) (+ relu)
// grid: (M/16), block: 256 threads (8 waves); wave w owns cols 16*w.
__global__ __launch_bounds__(256)
void gemm_wmma_f32(const float* __restrict__ A, const float* __restrict__ W,
                   const float* __restrict__ bias, float* __restrict__ out,
                   int relu) {
    __shared__ float sA[16 * 132];                 // padded stride: banks spread
    const int m0  = blockIdx.x * 16;
    const int tid = threadIdx.x;

    // stage 16x128 A tile once per block, fully coalesced
#pragma unroll
    for (int i = 0; i < 8; ++i) {
        int idx = i * 256 + tid;
        int r = idx >> 7, c = idx & 127;
        sA[r * 132 + c] = A[(size_t)(m0 + r) * DD + c];
    }
    __syncthreads();

    const int wave = tid >> 5;
    const int lane = tid & 31;
    const int n0   = wave * 16;
    const int row  = lane & 15;                    // M (for A) / N (for B) index
    const int half = lane >> 4;                    // 0: K=kb,kb+1   1: K=kb+2,kb+3
    const int koff = half * 2;

    v8f acc = {0.f, 0.f, 0.f, 0.f, 0.f, 0.f, 0.f, 0.f};
#pragma unroll
    for (int kb = 0; kb < DD; kb += 4) {
        v2f a, b;
        a.x = sA[row * 132 + kb + koff];
        a.y = sA[row * 132 + kb + koff + 1];
        b.x = W[(size_t)(kb + koff)     * DD + n0 + row];
        b.y = W[(size_t)(kb + koff + 1) * DD + n0 + row];
        acc = __builtin_amdgcn_wmma_f32_16x16x4_f32(
            false, a, false, b, (short)0, acc, false, false);
    }

    const int   col   = n0 + row;
    const float bv    = bias ? bias[col] : 0.0f;
    const int   rbase = m0 + half * 8;             // C/D: VGPR j -> M=j / M=j+8
#pragma unroll
    for (int j = 0; j < 8; ++j) {
        float v = acc[j] + bv;
        if (relu) v = fmaxf(v, 0.0f);
        out[(size_t)(rbase + j) * DD + col] = v;
    }
}

// ------------------------------------------- fused CSR gather + self + bias + relu
// One wave per node; lane owns a float4 slice of the 128-dim row.
// out[n] = relu( sum_{e in in(n)} hw[src_e]*dinv[src_e]*dinv[n] + hw[n]*dinv[n]^2 + b )
__global__ __launch_bounds__(256)
void gcn_gather(const float* __restrict__ hw, const int* __restrict__ rowptr,
                const int* __restrict__ colsrc, const float* __restrict__ dinv,
                const float* __restrict__ bias, float* __restrict__ out) {
    int n    = (int)((blockIdx.x * 256u + threadIdx.x) >> 5);
    int lane = threadIdx.x & 31;
    if (n >= NN) return;
    const float4* hp = (const float4*)hw;
    float dn = dinv[n];
    float4 hs = hp[(size_t)n * 32 + lane];
    float w0 = dn * dn;
    float ax = hs.x * w0, ay = hs.y * w0, az = hs.z * w0, aw = hs.w * w0;
    int e = rowptr[n], end = rowptr[n + 1];
    for (; e < end; ++e) {
        int s = colsrc[e];                          // wave-uniform
        float w = dinv[s] * dn;
        float4 hv = hp[(size_t)s * 32 + lane];      // coalesced 512B row
        ax += hv.x * w; ay += hv.y * w; az += hv.z * w; aw += hv.w * w;
    }
    float4 bv = ((const float4*)bias)[lane];
    float4 r;
    r.x = fmaxf(ax + bv.x, 0.f);
    r.y = fmaxf(ay + bv.y, 0.f);
    r.z = fmaxf(az + bv.z, 0.f);
    r.w = fmaxf(aw + bv.w, 0.f);
    ((float4*)out)[(size_t)n * 32 + lane] = r;
}

// --------------------------------------------------- pooling (batch is sorted)
__global__ void graph_bounds(const int* __restrict__ batch, int* __restrict__ gstart) {
    int g = blockIdx.x * blockDim.x + threadIdx.x;
    if (g > NG) return;
    if (g == NG) { gstart[NG] = NN; return; }
    int lo = 0, hi = NN;                            // lower_bound(batch, g)
    while (lo < hi) {
        int mid = (lo + hi) >> 1;
        if (batch[mid] < g) lo = mid + 1; else hi = mid;
    }
    gstart[g] = lo;
}

__global__ __launch_bounds__(128)
void pool_sorted(const float* __restrict__ h, const int* __restrict__ gstart,
                 float* __restrict__ G) {
    int g = blockIdx.x, d = threadIdx.x;
    int beg = gstart[g], end = gstart[g + 1];
    float acc = 0.f;
    for (int n = beg; n < end; ++n) acc += h[(size_t)n * DD + d];
    G[(size_t)g * DD + d] = acc;
}

// logits = Gh @ Wr2[128x10] + br2
__global__ void head2_kernel(const float* __restrict__ gh, const float* __restrict__ Wr2,
                             const float* __restrict__ br2, float* __restrict__ out) {
    int i = blockIdx.x * blockDim.x + threadIdx.x;
    if (i >= NG * NC) return;
    int g = i / NC, c = i % NC;
    float acc = br2[c];
#pragma unroll 8
    for (int k = 0; k < DD; ++k) acc += gh[(size_t)g * DD + k] * Wr2[k * NC + c];
    out[i] = acc;
}

// ---------------------------------------------------------------- launcher
extern "C" void kernel_launch(void* const* d_in, const int* in_sizes, int n_in,
                              void* d_out, int out_size, void* d_ws, size_t ws_size,
                              hipStream_t stream) {
    const float* x     = (const float*)d_in[0];
    const int*   src   = (const int*)d_in[1];   // c_2
    const int*   dst   = (const int*)d_in[2];   // u_2
    const int*   batch = (const int*)d_in[3];
    const float* W_enc = (const float*)d_in[4];
    const float* b_enc = (const float*)d_in[5];
    const float* Ws[3] = {(const float*)d_in[6], (const float*)d_in[8], (const float*)d_in[10]};
    const float* bs[3] = {(const float*)d_in[7], (const float*)d_in[9], (const float*)d_in[11]};
    const float* Wr1   = (const float*)d_in[12];
    const float* br1   = (const float*)d_in[13];
    const float* Wr2   = (const float*)d_in[14];
    const float* br2   = (const float*)d_in[15];

    char*  ws  = (char*)d_ws;
    size_t off = 0;
    auto carve = [&](size_t bytes) {
        void* p = ws + off;
        off += (bytes + 255) & ~(size_t)255;
        return p;
    };
    float* bufH   = (float*)carve((size_t)NN * DD * 4);  // node features h
    float* bufHW  = (float*)carve((size_t)NN * DD * 4);  // h @ W scratch
    int*   deg    = (int*)  carve((size_t)NN * 4);
    float* dinv   = (float*)carve((size_t)NN * 4);
    int*   rowptr = (int*)  carve((size_t)(NN + 1) * 4);
    int*   cursor = (int*)  carve((size_t)NN * 4);
    int*   colsrc = (int*)  carve((size_t)NE * 4);
    int*   bsum   = (int*)  carve((size_t)512 * 4);
    int*   gstart = (int*)  carve((size_t)(NG + 1) * 4);
    float* G      = (float*)carve((size_t)NG * DD * 4);
    float* Gh     = (float*)carve((size_t)NG * DD * 4);

    const int nscan = (NN + 255) / 256;   // 391 blocks <= 512

    // --- degree histogram over dst, dinv = rsqrt(deg+1)
    zero_f32<<<(NN / 4 + 255) / 256, 256, 0, stream>>>((float*)deg, NN / 4);
    deg_count<<<(NE + 255) / 256, 256, 0, stream>>>(dst, deg, NE);
    make_dinv<<<(NN + 255) / 256, 256, 0, stream>>>(deg, dinv, NN);

    // --- CSR build: rowptr = exclusive_scan(deg); colsrc[pos] = src
    scan_block<<<nscan, 256, 0, stream>>>(deg, rowptr, bsum, NN);
    scan_sums<<<1, 512, 0, stream>>>(bsum, nscan);
    add_offsets<<<nscan, 256, 0, stream>>>(rowptr, bsum, NN, NE);
    zero_f32<<<(NN / 4 + 255) / 256, 256, 0, stream>>>((float*)cursor, NN / 4);
    csr_fill<<<(NE + 255) / 256, 256, 0, stream>>>(src, dst, rowptr, cursor, colsrc, NE);

    // --- graph boundaries for sorted batch
    graph_bounds<<<(NG + 1 + 255) / 256, 256, 0, stream>>>(batch, gstart);

    // --- encoder: h = x @ W_enc + b_enc
    gemm_wmma_f32<<<NN / 16, 256, 0, stream>>>(x, W_enc, b_enc, bufH, 0);

    // --- 3 GCN layers: GEMM -> fused gather/self/bias/relu
    for (int l = 0; l < 3; ++l) {
        gemm_wmma_f32<<<NN / 16, 256, 0, stream>>>(bufH, Ws[l], nullptr, bufHW, 0);
        gcn_gather<<<(NN * 32 + 255) / 256, 256, 0, stream>>>(bufHW, rowptr, colsrc,
                                                              dinv, bs[l], bufH);
    }

    // --- pooling (segmented reduction, batch sorted)
    pool_sorted<<<NG, 128, 0, stream>>>(bufH, gstart, G);

    // --- head: relu(G @ Wr1 + br1) @ Wr2 + br2
    gemm_wmma_f32<<<NG / 16, 256, 0, stream>>>(G, Wr1, br1, Gh, 1);
    head2_kernel<<<(NG * NC + 255) / 256, 256, 0, stream>>>(Gh, Wr2, br2, (float*)d_out);
}